// MultiTaskModel_5746666242651
// MI455X (gfx1250) — compile-verified
//
#include <hip/hip_runtime.h>

typedef float v2f __attribute__((ext_vector_type(2)));
typedef float v4f __attribute__((ext_vector_type(4)));
typedef float v8f __attribute__((ext_vector_type(8)));

#define IN_FEAT   1536
#define SEG_HID   256
#define NUM_DR    5
#define NUM_ED    3
#define NUM_SEG   5
#define BATCH     32
#define PLANE     (512 * 512)

// ---------------------------------------------------------------------------
// Pack W_dr [1536,5] + W_ed [1536,3] into Bpad1 [1536,16] (cols 0-4 dr, 5-7 ed,
// rest zero), and W2 [256,5] into Bpad2 [256,16] (cols 0-4, rest zero).
// ---------------------------------------------------------------------------
__global__ __launch_bounds__(256) void pack_weights_kernel(
    const float* __restrict__ W_dr, const float* __restrict__ W_ed,
    const float* __restrict__ W2, float* __restrict__ Bpad1,
    float* __restrict__ Bpad2) {
  int idx = blockIdx.x * blockDim.x + threadIdx.x;
  if (idx < IN_FEAT * 16) {
    int k = idx >> 4, n = idx & 15;
    float v = 0.0f;
    if (n < NUM_DR)               v = W_dr[k * NUM_DR + n];
    else if (n < NUM_DR + NUM_ED) v = W_ed[k * NUM_ED + (n - NUM_DR)];
    Bpad1[idx] = v;
  } else {
    int idx2 = idx - IN_FEAT * 16;
    if (idx2 < SEG_HID * 16) {
      int k = idx2 >> 4, n = idx2 & 15;
      Bpad2[idx2] = (n < NUM_SEG) ? W2[k * NUM_SEG + n] : 0.0f;
    }
  }
}

// ---------------------------------------------------------------------------
// h = relu(F @ W1 + b1)   [32,1536] x [1536,256] -> [32,256]
// One 16x16 tile per wave; 32 tiles = 4 blocks x 8 waves. K-step 4 via
// V_WMMA_F32_16X16X4_F32 (fp32 in, fp32 acc -> matches fp32 reference).
// ---------------------------------------------------------------------------
__global__ __launch_bounds__(256) void gemm_h_kernel(
    const float* __restrict__ F, const float* __restrict__ W1,
    const float* __restrict__ b1, float* __restrict__ h) {
  const int lane  = threadIdx.x & 31;
  const int wave  = threadIdx.x >> 5;
  const int tile  = blockIdx.x * 8 + wave;   // 0..31
  const int mTile = (tile & 1) * 16;         // 0 or 16
  const int nTile = (tile >> 1) * 16;        // 0..240
  const int half  = lane >> 4;               // 0: K=0,1  1: K=2,3
  const int idx16 = lane & 15;               // M for A-lanes, N for B-lanes
  const int kb    = half * 2;

  const float* __restrict__ Arow = F + (size_t)(mTile + idx16) * IN_FEAT + kb;
  const float* __restrict__ Bcol = W1 + nTile + idx16;

  v8f acc = {};
#pragma unroll 8
  for (int k = 0; k < IN_FEAT; k += 4) {
    v2f a, b;
    a.x = Arow[k];
    a.y = Arow[k + 1];
    b.x = Bcol[(size_t)(k + kb) * SEG_HID];
    b.y = Bcol[(size_t)(k + kb + 1) * SEG_HID];
    acc = __builtin_amdgcn_wmma_f32_16x16x4_f32(false, a, false, b,
                                                (short)0, acc, false, false);
  }

  // D layout: VGPR r -> M = r + half*8, N = lane&15
  const float bias = b1[nTile + idx16];
  const int mbase  = mTile + half * 8;
#pragma unroll
  for (int r = 0; r < 8; ++r) {
    float v = acc[r] + bias;
    v = v > 0.0f ? v : 0.0f;
    h[(size_t)(mbase + r) * SEG_HID + nTile + idx16] = v;
  }
}

// ---------------------------------------------------------------------------
// Heads: D = F @ Bpad1  [32,1536] x [1536,16]; cols 0-4 -> dr_out (+b_dr),
// cols 5-7 -> edema_out (+b_ed). 2 waves (one per 16-row tile).
// d_out layout: [0,160) dr flat, [160,256) edema flat.
// ---------------------------------------------------------------------------
__global__ __launch_bounds__(64) void gemm_heads_kernel(
    const float* __restrict__ F, const float* __restrict__ Bpad1,
    const float* __restrict__ b_dr, const float* __restrict__ b_ed,
    float* __restrict__ out) {
  const int lane  = threadIdx.x & 31;
  const int wave  = threadIdx.x >> 5;
  const int mTile = wave * 16;
  const int half  = lane >> 4;
  const int idx16 = lane & 15;
  const int kb    = half * 2;

  const float* __restrict__ Arow = F + (size_t)(mTile + idx16) * IN_FEAT + kb;
  const float* __restrict__ Bcol = Bpad1 + idx16;

  v8f acc = {};
#pragma unroll 8
  for (int k = 0; k < IN_FEAT; k += 4) {
    v2f a, b;
    a.x = Arow[k];
    a.y = Arow[k + 1];
    b.x = Bcol[(k + kb) * 16];
    b.y = Bcol[(k + kb + 1) * 16];
    acc = __builtin_amdgcn_wmma_f32_16x16x4_f32(false, a, false, b,
                                                (short)0, acc, false, false);
  }

  const int n     = idx16;
  const int mbase = mTile + half * 8;
#pragma unroll
  for (int r = 0; r < 8; ++r) {
    const int m = mbase + r;
    if (n < NUM_DR) {
      out[m * NUM_DR + n] = acc[r] + b_dr[n];
    } else if (n < NUM_DR + NUM_ED) {
      out[BATCH * NUM_DR + m * NUM_ED + (n - NUM_DR)] = acc[r] + b_ed[n - NUM_DR];
    }
  }
}

// ---------------------------------------------------------------------------
// seg_vec = h @ Bpad2 + b2   [32,256] x [256,16] -> keep cols 0-4. 2 waves.
// ---------------------------------------------------------------------------
__global__ __launch_bounds__(64) void gemm_seg_kernel(
    const float* __restrict__ h, const float* __restrict__ Bpad2,
    const float* __restrict__ b2, float* __restrict__ segv) {
  const int lane  = threadIdx.x & 31;
  const int wave  = threadIdx.x >> 5;
  const int mTile = wave * 16;
  const int half  = lane >> 4;
  const int idx16 = lane & 15;
  const int kb    = half * 2;

  const float* __restrict__ Arow = h + (size_t)(mTile + idx16) * SEG_HID + kb;
  const float* __restrict__ Bcol = Bpad2 + idx16;

  v8f acc = {};
#pragma unroll 8
  for (int k = 0; k < SEG_HID; k += 4) {
    v2f a, b;
    a.x = Arow[k];
    a.y = Arow[k + 1];
    b.x = Bcol[(k + kb) * 16];
    b.y = Bcol[(k + kb + 1) * 16];
    acc = __builtin_amdgcn_wmma_f32_16x16x4_f32(false, a, false, b,
                                                (short)0, acc, false, false);
  }

  const int n     = idx16;
  const int mbase = mTile + half * 8;
  if (n < NUM_SEG) {
    const float bias = b2[n];
#pragma unroll
    for (int r = 0; r < 8; ++r) {
      segv[(mbase + r) * NUM_SEG + n] = acc[r] + bias;
    }
  }
}

// ---------------------------------------------------------------------------
// seg_out broadcast: 160 planes (m*5+c) of 512*512 floats, each a constant.
// 167.8 MB of pure streaming stores -> B128 nontemporal. grid (64, 160),
// 256 thr/block, 4 x float4 per thread = 16KB per block.
// ---------------------------------------------------------------------------
__global__ __launch_bounds__(256) void broadcast_seg_kernel(
    const float* __restrict__ segv, float* __restrict__ out) {
  const int plane = blockIdx.y;              // 0..159
  const float v   = segv[plane];
  v4f vv = {v, v, v, v};
  v4f* __restrict__ dst = (v4f*)(out + (size_t)plane * PLANE)
                          + (size_t)blockIdx.x * 1024 + threadIdx.x;
#pragma unroll
  for (int i = 0; i < 4; ++i) {
    __builtin_nontemporal_store(vv, dst + i * 256);
  }
}

// ---------------------------------------------------------------------------
extern "C" void kernel_launch(void* const* d_in, const int* in_sizes, int n_in,
                              void* d_out, int out_size, void* d_ws, size_t ws_size,
                              hipStream_t stream) {
  const float* F    = (const float*)d_in[0];
  const float* W_dr = (const float*)d_in[1];
  const float* b_dr = (const float*)d_in[2];
  const float* W_ed = (const float*)d_in[3];
  const float* b_ed = (const float*)d_in[4];
  const float* W1   = (const float*)d_in[5];
  const float* b1   = (const float*)d_in[6];
  const float* W2   = (const float*)d_in[7];
  const float* b2   = (const float*)d_in[8];
  float* out = (float*)d_out;
  float* ws  = (float*)d_ws;

  float* Bpad1 = ws;                          // 1536*16 = 24576 floats
  float* Bpad2 = ws + 24576;                  //  256*16 =  4096 floats
  float* hbuf  = ws + 24576 + 4096;           //   32*256 = 8192 floats
  float* segv  = ws + 24576 + 4096 + 8192;    //   32*5  =   160 floats

  // (1536*16 + 256*16) = 28672 threads = 112 blocks of 256
  pack_weights_kernel<<<112, 256, 0, stream>>>(W_dr, W_ed, W2, Bpad1, Bpad2);
  gemm_h_kernel<<<4, 256, 0, stream>>>(F, W1, b1, hbuf);
  gemm_heads_kernel<<<1, 64, 0, stream>>>(F, Bpad1, b_dr, b_ed, out);
  gemm_seg_kernel<<<1, 64, 0, stream>>>(hbuf, Bpad2, b2, segv);
  // seg_out starts after 160 + 96 = 256 floats (16B-aligned)
  broadcast_seg_kernel<<<dim3(64, 160), 256, 0, stream>>>(segv, out + 256);
}